// GATWithJK_12197707120917
// MI455X (gfx1250) — compile-verified
//
#include <hip/hip_runtime.h>
#include <hip/hip_bf16.h>
#include <math.h>

// ---------------- problem constants (match reference) ----------------
#define NN     50000        // nodes; NOTE: 50000 = 3125 * 16 -> exact M tiling
#define EE     800000
#define GG     500
#define INC    128
#define HIDC   32
#define NHEAD  4
#define CC     128          // HID*HEADS
#define LL     3
#define HLS    192          // (L*C)/2
#define G4     768          // 4*HL
#define OUTC   2
#define NEGS   0.2f
#define ETOT   (EE + NN)    // edges + self loops

typedef float v2f __attribute__((ext_vector_type(2)));
typedef float v8f __attribute__((ext_vector_type(8)));

// ---------------- utility kernels ----------------
__global__ __launch_bounds__(256) void zero_f32(float* p, long n) {
    long i = (long)blockIdx.x * blockDim.x + threadIdx.x;
    long stride = (long)gridDim.x * blockDim.x;
    for (; i < n; i += stride) p[i] = 0.0f;
}

// ---------- fp32 WMMA GEMM: C[M x Ncol] = A[M x K] * B (+C) -----------
// EXACT TILING REQUIRED: M % 16 == 0, Ncol % (NT*16) == 0, K % 4 == 0.
// (All call sites satisfy this: M=50000=3125*16, Ncol in {128,768},
//  K in {128,192}.)  No bounds guards -> no exec-mask churn around WMMA.
// One wave computes a 16 x (NT*16) strip: the A fragment (16x4, one v2f per
// lane) is loaded once per K-step and reused across NT accumulators, so the
// inner loop issues 1 A load + NT B loads + NT v_wmma_f32_16x16x4_f32.
// BT=false : B stored [K x Ncol] row-major (ldb = Ncol)
// BT=true  : B stored [Ncol x K] row-major (ldb = K)  => uses B^T
// ACC      : accumulate into existing C
// Fragment layouts per CDNA5 ISA 7.12.2:
//   A (16x4 f32):  lane l -> row m = l%16;  VGPR v holds K = k0 + 2*(l/16) + v
//   B (4x16 f32):  lane l -> col n = l%16;  VGPR v holds K = k0 + 2*(l/16) + v
//   C/D (16x16):   VGPR r -> row m = r + 8*(l/16), col n = l%16
template<bool BT, bool ACC, int NT>
__global__ __launch_bounds__(32)
void wmma_gemm_f32(const float* __restrict__ A, const float* __restrict__ B,
                   float* __restrict__ Cm, int K, int lda, int ldb, int ldc)
{
    const int tn0  = blockIdx.x * NT;      // first 16-wide tile col of strip
    const int tm   = blockIdx.y;           // tile row
    const int lane = threadIdx.x;
    const int half = lane >> 4;
    const int lm   = lane & 15;

    const int arow = tm * 16 + lm;         // A row for this lane
    // per-lane C base: row (tm*16 + 8*half), col (tn0*16 + lm)
    float* __restrict__ cbase =
        Cm + (size_t)(tm * 16 + 8 * half) * ldc + tn0 * 16 + lm;

    v8f acc[NT];
#pragma unroll
    for (int j = 0; j < NT; ++j) {
#pragma unroll
        for (int r = 0; r < 8; ++r)
            acc[j][r] = ACC ? cbase[(size_t)r * ldc + j * 16] : 0.0f;
    }

    const float* __restrict__ abase = A + (size_t)arow * lda;
    for (int k0 = 0; k0 < K; k0 += 4) {
        const int ka = k0 + 2 * half;               // even -> 8B aligned
        v2f a = *reinterpret_cast<const v2f*>(&abase[ka]);
#pragma unroll
        for (int j = 0; j < NT; ++j) {
            const int bcol = (tn0 + j) * 16 + lm;
            v2f b;
            if (BT) {
                b = *reinterpret_cast<const v2f*>(&B[(size_t)bcol * ldb + ka]);
            } else {
                b[0] = B[(size_t)ka * ldb + bcol];
                b[1] = B[(size_t)(ka + 1) * ldb + bcol];
            }
            // 8-arg form: (neg_a, A, neg_b, B, c_mod, C, reuse_a, reuse_b)
            acc[j] = __builtin_amdgcn_wmma_f32_16x16x4_f32(
                false, a, false, b, (short)0, acc[j], false, false);
        }
    }

#pragma unroll
    for (int j = 0; j < NT; ++j) {
#pragma unroll
        for (int r = 0; r < 8; ++r)
            cbase[(size_t)r * ldc + j * 16] = acc[j][r];
    }
}

// ---------------- GAT attention helpers ----------------
__device__ __forceinline__ unsigned fenc(float f) {
    unsigned u = __float_as_uint(f);
    return (u & 0x80000000u) ? ~u : (u | 0x80000000u);
}
__device__ __forceinline__ float fdec(unsigned u) {
    unsigned v = (u & 0x80000000u) ? (u & 0x7FFFFFFFu) : ~u;
    return __uint_as_float(v);
}
__device__ __forceinline__ float lrelu(float x) {
    return x > 0.0f ? x : NEGS * x;
}
__device__ __forceinline__ void edge_sd(const int* __restrict__ ei, int e,
                                        int& s, int& d) {
    if (e < EE) { s = ei[e]; d = ei[EE + e]; }
    else        { s = e - EE; d = e - EE; }      // self loop
}

// es[n,h] = sum_k hp[n,h,k]*as[h,k];  ed likewise
__global__ __launch_bounds__(256)
void att_coeffs(const float* __restrict__ hp, const float* __restrict__ as_,
                const float* __restrict__ ad_, float* __restrict__ es,
                float* __restrict__ ed)
{
    int gid = blockIdx.x * blockDim.x + threadIdx.x;
    if (gid >= NN * NHEAD) return;
    int node = gid / NHEAD, h = gid % NHEAD;
    const float* hv = hp + (size_t)node * CC + h * HIDC;
    const float* av = as_ + h * HIDC;
    const float* dv = ad_ + h * HIDC;
    float s = 0.0f, d = 0.0f;
#pragma unroll
    for (int k = 0; k < HIDC; ++k) { float v = hv[k]; s += v * av[k]; d += v * dv[k]; }
    es[gid] = s; ed[gid] = d;
}

// pass 1: segment max of leaky(es[src]+ed[dst]) keyed by dst (monotone-uint atomicMax)
__global__ __launch_bounds__(256)
void edge_max(const int* __restrict__ ei, const float* __restrict__ es,
              const float* __restrict__ ed, unsigned* __restrict__ mx)
{
    int gid = blockIdx.x * blockDim.x + threadIdx.x;
    if (gid >= ETOT * NHEAD) return;
    int e = gid >> 2, h = gid & 3, s, d;
    edge_sd(ei, e, s, d);
    float v = lrelu(es[s * NHEAD + h] + ed[d * NHEAD + h]);
    atomicMax(&mx[d * NHEAD + h], fenc(v));
}

// pass 2: ssum[dst] += exp(e - max[dst])
__global__ __launch_bounds__(256)
void edge_expsum(const int* __restrict__ ei, const float* __restrict__ es,
                 const float* __restrict__ ed, const unsigned* __restrict__ mx,
                 float* __restrict__ ssum)
{
    int gid = blockIdx.x * blockDim.x + threadIdx.x;
    if (gid >= ETOT * NHEAD) return;
    int e = gid >> 2, h = gid & 3, s, d;
    edge_sd(ei, e, s, d);
    float v = lrelu(es[s * NHEAD + h] + ed[d * NHEAD + h]);
    float ex = expf(v - fdec(mx[d * NHEAD + h]));
    atomicAdd(&ssum[d * NHEAD + h], ex);
}

// pass 3: out[dst,h,:] += h[src,h,:] * alpha
__global__ __launch_bounds__(256)
void edge_scatter(const int* __restrict__ ei, const float* __restrict__ es,
                  const float* __restrict__ ed, const unsigned* __restrict__ mx,
                  const float* __restrict__ ssum, const float* __restrict__ hp,
                  float* __restrict__ out)
{
    int gid = blockIdx.x * blockDim.x + threadIdx.x;
    if (gid >= ETOT * NHEAD) return;
    int e = gid >> 2, h = gid & 3, s, d;
    edge_sd(ei, e, s, d);
    float v = lrelu(es[s * NHEAD + h] + ed[d * NHEAD + h]);
    float ex = expf(v - fdec(mx[d * NHEAD + h]));
    float alpha = ex / (ssum[d * NHEAD + h] + 1e-16f);
    const float* hv = hp + (size_t)s * CC + h * HIDC;
    float* ov = out + (size_t)d * CC + h * HIDC;
#pragma unroll
    for (int k = 0; k < HIDC; ++k) atomicAdd(&ov[k], hv[k] * alpha);
}

__global__ __launch_bounds__(256)
void bias_relu(float* __restrict__ x, const float* __restrict__ b)
{
    int gid = blockIdx.x * blockDim.x + threadIdx.x;
    if (gid >= NN * CC) return;
    int c = gid % CC;
    float v = x[gid] + b[c];
    x[gid] = v > 0.0f ? v : 0.0f;
}

// ---------------- LSTM gates (torch order i,f,g,o) ----------------
__device__ __forceinline__ float sigf(float x) { return 1.0f / (1.0f + expf(-x)); }

__global__ __launch_bounds__(256)
void lstm_gates(const float* __restrict__ z, const float* __restrict__ bih,
                const float* __restrict__ bhh, float* __restrict__ h,
                float* __restrict__ c, float* __restrict__ hs_out)
{
    int gid = blockIdx.x * blockDim.x + threadIdx.x;
    if (gid >= NN * HLS) return;
    int node = gid / HLS, j = gid % HLS;
    const float* zr = z + (size_t)node * G4;
    float zi = zr[j]           + bih[j]           + bhh[j];
    float zf = zr[HLS + j]     + bih[HLS + j]     + bhh[HLS + j];
    float zg = zr[2 * HLS + j] + bih[2 * HLS + j] + bhh[2 * HLS + j];
    float zo = zr[3 * HLS + j] + bih[3 * HLS + j] + bhh[3 * HLS + j];
    float cc = sigf(zf) * c[gid] + sigf(zi) * tanhf(zg);
    float hh = sigf(zo) * tanhf(cc);
    c[gid] = cc; h[gid] = hh; hs_out[gid] = hh;
}

// ---------------- JK attention + weighted sum ----------------
__global__ __launch_bounds__(256)
void jk_attn(const float* __restrict__ hsf, const float* __restrict__ hsb,
             const float* __restrict__ Wa, const float* __restrict__ ba,
             const float* __restrict__ xs, float* __restrict__ jk)
{
    int node = blockIdx.x * blockDim.x + threadIdx.x;
    if (node >= NN) return;
    float sc[LL];
#pragma unroll
    for (int l = 0; l < LL; ++l) {
        const float* f = hsf + ((size_t)l * NN + node) * HLS;
        const float* b = hsb + ((size_t)l * NN + node) * HLS;
        float s = ba[0];
        for (int j = 0; j < HLS; ++j) s += f[j] * Wa[j];
        for (int j = 0; j < HLS; ++j) s += b[j] * Wa[HLS + j];
        sc[l] = s;
    }
    float m = fmaxf(sc[0], fmaxf(sc[1], sc[2]));
    float e0 = expf(sc[0] - m), e1 = expf(sc[1] - m), e2 = expf(sc[2] - m);
    float inv = 1.0f / (e0 + e1 + e2);
    float a0 = e0 * inv, a1 = e1 * inv, a2 = e2 * inv;
    const float* x0 = xs + (size_t)node * CC;
    const float* x1 = xs + ((size_t)NN + node) * CC;
    const float* x2 = xs + ((size_t)2 * NN + node) * CC;
    float* o = jk + (size_t)node * CC;
#pragma unroll 4
    for (int cidx = 0; cidx < CC; ++cidx)
        o[cidx] = x0[cidx] * a0 + x1[cidx] * a1 + x2[cidx] * a2;
}

// ---------------- mean pool + final linear ----------------
__global__ __launch_bounds__(256)
void pool_scatter(const int* __restrict__ batch, const float* __restrict__ jk,
                  float* __restrict__ pooled, float* __restrict__ counts)
{
    int node = blockIdx.x * blockDim.x + threadIdx.x;
    if (node >= NN) return;
    int g = batch[node];
    atomicAdd(&counts[g], 1.0f);
    const float* v = jk + (size_t)node * CC;
    float* p = pooled + (size_t)g * CC;
    for (int cidx = 0; cidx < CC; ++cidx) atomicAdd(&p[cidx], v[cidx]);
}

__global__ __launch_bounds__(64)
void final_linear(const float* __restrict__ pooled, const float* __restrict__ counts,
                  const float* __restrict__ Wl, const float* __restrict__ bl,
                  float* __restrict__ out)
{
    int gid = blockIdx.x * blockDim.x + threadIdx.x;
    if (gid >= GG * OUTC) return;
    int g = gid / OUTC, o = gid % OUTC;
    float inv = 1.0f / fmaxf(counts[g], 1.0f);
    const float* p = pooled + (size_t)g * CC;
    float acc = bl[o];
    for (int cidx = 0; cidx < CC; ++cidx) acc += p[cidx] * inv * Wl[cidx * OUTC + o];
    out[gid] = acc;
}

// ---------------- host orchestration ----------------
static inline int cdiv(long a, int b) { return (int)((a + b - 1) / b); }

extern "C" void kernel_launch(void* const* d_in, const int* in_sizes, int n_in,
                              void* d_out, int out_size, void* d_ws, size_t ws_size,
                              hipStream_t stream) {
    const float* x   = (const float*)d_in[0];
    const int*   ei  = (const int*)d_in[1];
    const int*   bat = (const int*)d_in[2];
    const float* W[3]  = { (const float*)d_in[3],  (const float*)d_in[7],  (const float*)d_in[11] };
    const float* as_[3]= { (const float*)d_in[4],  (const float*)d_in[8],  (const float*)d_in[12] };
    const float* ad_[3]= { (const float*)d_in[5],  (const float*)d_in[9],  (const float*)d_in[13] };
    const float* bb[3] = { (const float*)d_in[6],  (const float*)d_in[10], (const float*)d_in[14] };
    const float* Wih[2] = { (const float*)d_in[15], (const float*)d_in[19] };
    const float* Whh[2] = { (const float*)d_in[16], (const float*)d_in[20] };
    const float* bih[2] = { (const float*)d_in[17], (const float*)d_in[21] };
    const float* bhh[2] = { (const float*)d_in[18], (const float*)d_in[22] };
    const float* Wa = (const float*)d_in[23];
    const float* ba = (const float*)d_in[24];
    const float* Wl = (const float*)d_in[25];
    const float* bl = (const float*)d_in[26];
    float* out = (float*)d_out;

    // workspace layout (floats)
    float* w = (float*)d_ws;
    size_t off = 0;
    float* XS  = w + off; off += (size_t)LL * NN * CC;   // layer outputs xs[0..2]
    float* HP  = w + off; off += (size_t)NN * CC;        // projected features
    float* ES  = w + off; off += (size_t)NN * NHEAD;
    float* ED  = w + off; off += (size_t)NN * NHEAD;
    float* MXf = w + off; off += (size_t)NN * NHEAD;     // encoded-uint max
    float* SS  = w + off; off += (size_t)NN * NHEAD;
    float* Z   = w + off; off += (size_t)NN * G4;        // LSTM pre-activations
    float* HSt = w + off; off += (size_t)NN * HLS;
    float* CSt = w + off; off += (size_t)NN * HLS;
    float* HSF = w + off; off += (size_t)LL * NN * HLS;
    float* HSB = w + off; off += (size_t)LL * NN * HLS;
    float* JK  = w + off; off += (size_t)NN * CC;
    float* PO  = w + off; off += (size_t)GG * CC;
    float* CT  = w + off; off += (size_t)GG;
    unsigned* MX = (unsigned*)MXf;
    (void)ws_size; (void)n_in; (void)in_sizes; (void)out_size;

    const int ZB = 1024;  // blocks for grid-stride zero fill
    dim3 b256(256);
    constexpr int NT = 8;                 // 16 x 128 strip per wave

    // -------- GAT layers --------
    for (int l = 0; l < LL; ++l) {
        const float* hin = (l == 0) ? x : (XS + (size_t)(l - 1) * NN * CC);
        float* xsl = XS + (size_t)l * NN * CC;

        // projection: HP = hin @ W[l]   (M=N=50000, Ncol=128, K=128)
        wmma_gemm_f32<false, false, NT><<<dim3(CC / (16 * NT), NN / 16), 32, 0, stream>>>(
            hin, W[l], HP, INC, INC, CC, CC);

        att_coeffs<<<cdiv((long)NN * NHEAD, 256), b256, 0, stream>>>(HP, as_[l], ad_[l], ES, ED);

        zero_f32<<<ZB, b256, 0, stream>>>(MXf, (long)NN * NHEAD);  // encoded 0 == ordered floor
        zero_f32<<<ZB, b256, 0, stream>>>(SS,  (long)NN * NHEAD);
        zero_f32<<<ZB, b256, 0, stream>>>(xsl, (long)NN * CC);

        int egrid = cdiv((long)ETOT * NHEAD, 256);
        edge_max    <<<egrid, b256, 0, stream>>>(ei, ES, ED, MX);
        edge_expsum <<<egrid, b256, 0, stream>>>(ei, ES, ED, MX, SS);
        edge_scatter<<<egrid, b256, 0, stream>>>(ei, ES, ED, MX, SS, HP, xsl);

        bias_relu<<<cdiv((long)NN * CC, 256), b256, 0, stream>>>(xsl, bb[l]);
    }

    // -------- Bi-LSTM over the L=3 layer outputs --------
    dim3 zgrid(G4 / (16 * NT), NN / 16);   // 6 x 3125 strips
    for (int dir = 0; dir < 2; ++dir) {
        zero_f32<<<ZB, b256, 0, stream>>>(HSt, (long)NN * HLS);
        zero_f32<<<ZB, b256, 0, stream>>>(CSt, (long)NN * HLS);
        for (int step = 0; step < LL; ++step) {
            int t = (dir == 0) ? step : (LL - 1 - step);
            const float* xt = XS + (size_t)t * NN * CC;
            float* hsOut = ((dir == 0) ? HSF : HSB) + (size_t)t * NN * HLS;
            // Z = xt @ Wih^T          (B stored [768 x 128], BT)
            wmma_gemm_f32<true, false, NT><<<zgrid, 32, 0, stream>>>(
                xt, Wih[dir], Z, INC, INC, INC, G4);
            // Z += h @ Whh^T          (B stored [768 x 192], BT)
            wmma_gemm_f32<true, true, NT><<<zgrid, 32, 0, stream>>>(
                HSt, Whh[dir], Z, HLS, HLS, HLS, G4);
            lstm_gates<<<cdiv((long)NN * HLS, 256), b256, 0, stream>>>(
                Z, bih[dir], bhh[dir], HSt, CSt, hsOut);
        }
    }

    // -------- JK attention --------
    jk_attn<<<cdiv(NN, 256), b256, 0, stream>>>(HSF, HSB, Wa, ba, XS, JK);

    // -------- mean pool + classifier --------
    zero_f32<<<cdiv((long)GG * CC, 256), b256, 0, stream>>>(PO, (long)GG * CC);
    zero_f32<<<1, b256, 0, stream>>>(CT, (long)GG);
    pool_scatter<<<cdiv(NN, 256), b256, 0, stream>>>(bat, JK, PO, CT);
    final_linear<<<cdiv((long)GG * OUTC, 64), dim3(64), 0, stream>>>(PO, CT, Wl, bl, out);
}